// KNO2d_8675833938157
// MI455X (gfx1250) — compile-verified
//
#include <hip/hip_runtime.h>
#include <hip/hip_bf16.h>
#include <math.h>
#include <stdint.h>

// ---------------- problem dims ----------------
#define BN 8
#define TN 10
#define CN 2
#define OPN 32
#define BCN 16           // B*C
#define NPLANE 512       // BC*OP
#define PSZ 65536        // 256*256
#define NPIX 1048576     // BC*256*256
#define OUTSZ 10485760   // B*T*C*H*W

// ---------------- workspace layout (float offsets, ~309 MB total) ----
#define OFF_H0   ((size_t)0)                      // 512*65536
#define OFF_H6   ((size_t)33554432)               // 512*65536
#define OFF_GR   ((size_t)67108864)               // 512*4096
#define OFF_GI   (OFF_GR + 2097152)
#define OFF_TR   (OFF_GI + 2097152)               // 512*512
#define OFF_TI   (OFF_TR + 262144)
#define OFF_UR   (OFF_TI + 262144)
#define OFF_UI   (OFF_UR + 262144)
#define OFF_Z    (OFF_UI + 262144)                // 512*8192 (interleaved re/im)
#define OFF_ER   (OFF_Z  + 4194304)               // 256*1024
#define OFF_EI   (OFF_ER + 262144)
#define OFF_FW   (OFF_EI + 262144)                // 256*32
#define OFF_FHC  (OFF_FW   + 8192)                // 32*256
#define OFF_FHS  (OFF_FHC  + 8192)
#define OFF_FHNS (OFF_FHS  + 8192)
#define OFF_IFC  (OFF_FHNS + 8192)                // 256*32
#define OFF_IFS  (OFF_IFC  + 8192)
#define OFF_IFNS (OFF_IFS  + 8192)
#define OFF_ITAB (OFF_IFNS + 8192)                // 32*256

// ---------------- WMMA helpers (V_WMMA_F32_16X16X4_F32, wave32) ------
typedef __attribute__((ext_vector_type(2))) float v2f;
typedef __attribute__((ext_vector_type(8))) float v8f;
typedef __attribute__((ext_vector_type(4))) unsigned int u32x4;
typedef __attribute__((ext_vector_type(4))) int i32x4;
typedef __attribute__((ext_vector_type(8))) int i32x8;

__device__ __forceinline__ v8f wmma4(v2f a, v2f b, v8f c) {
  // 8 args: (neg_a, A, neg_b, B, c_mod, C, reuse_a, reuse_b)
  return __builtin_amdgcn_wmma_f32_16x16x4_f32(false, a, false, b, (short)0, c,
                                               false, false);
}
// A is MxK row-major, 16x4 tile at (m0,k0): lane<16 -> K={k0,k0+1}, lane>=16 -> K={k0+2,k0+3}
__device__ __forceinline__ v2f ldA(const float* A, int lda, int m0, int k0, int lane) {
  const float* p = A + (size_t)(m0 + (lane & 15)) * lda + k0 + 2 * (lane >> 4);
  v2f r; r.x = p[0]; r.y = p[1]; return r;
}
// B is KxN row-major, 4x16 tile at (k0,n0)
__device__ __forceinline__ v2f ldB(const float* B, int ldb, int k0, int n0, int lane) {
  const float* p = B + (size_t)(k0 + 2 * (lane >> 4)) * ldb + n0 + (lane & 15);
  v2f r; r.x = p[0]; r.y = p[ldb]; return r;
}

// ---------------- Tensor Data Mover: 2D fp32 tile -> LDS -------------
// Builds a D# per CDNA5 ISA ch.8: group0 = {count|flags, lds_addr,
// global_addr[56:0], type=2}; group1 = {data_size=4B, tensor_dim0/1,
// tile_dim0/1, dim0_stride}. Tracked by TENSORcnt.
__device__ __forceinline__ void tdm_load_2d(unsigned lds_byte_off,
                                            const void* gptr,
                                            unsigned tile_w, unsigned tile_h,
                                            unsigned stride_elems) {
  unsigned long long ga = (unsigned long long)(uintptr_t)gptr;
  u32x4 g0;
  g0[0] = 1u;                                          // count=1 (valid, user mode)
  g0[1] = lds_byte_off;                                // LDS dest (bytes)
  g0[2] = (unsigned)(ga & 0xffffffffu);                // global_addr[31:0]
  g0[3] = (unsigned)((ga >> 32) & 0x01ffffffu) | (2u << 30);  // addr[56:32] | type=2
  i32x8 g1;
  g1[0] = (int)(2u << 16);                             // workgroup_mask=0, data_size=4B
  g1[1] = (int)((tile_w & 0xffffu) << 16);             // tensor_dim0[15:0] (== tile_w)
  g1[2] = (int)((tile_w >> 16) | ((tile_h & 0xffffu) << 16)); // dim0 hi | tensor_dim1 lo
  g1[3] = (int)((tile_h >> 16) | (tile_w << 16));      // dim1 hi | tile_dim0
  g1[4] = (int)tile_h;                                 // tile_dim1 | tile_dim2=0
  g1[5] = (int)stride_elems;                           // tensor_dim0_stride[31:0]
  g1[6] = 0;                                           // stride hi | dim1_stride lo
  g1[7] = 0;
  i32x4 z4 = {0, 0, 0, 0};
#if __clang_major__ >= 23
  i32x8 z8 = {0, 0, 0, 0, 0, 0, 0, 0};
  __builtin_amdgcn_tensor_load_to_lds(g0, g1, z4, z4, z8, 0);
#else
  __builtin_amdgcn_tensor_load_to_lds(g0, g1, z4, z4, 0);
#endif
}

// ---------------- twiddle / coefficient tables -----------------------
__global__ void k_tables(float* fw, float* fhc, float* fhs, float* fhns,
                         float* ifc, float* ifs, float* ifns, float* itab) {
  int idx = blockIdx.x * blockDim.x + threadIdx.x;
  if (idx >= 8192) return;
  const float STEP = 6.283185307179586f / 256.0f;
  { // forward col DFT: fw[ww][2y+ri], e^{-i 2pi y ww/256}: (cos, -sin)
    int ww = idx >> 5, n = idx & 31, y = n >> 1;
    float th = STEP * (float)((y * ww) & 255);
    fw[idx] = (n & 1) ? -sinf(th) : cosf(th);
  }
  { // forward row DFT to 32 kept rows: fh*[j][hh]
    int j = idx >> 8, hh = idx & 255;
    int x = (j < 16) ? j : (224 + j);           // rows 0..15 and 240..255
    float th = STEP * (float)((x * hh) & 255);
    float c = cosf(th), s = sinf(th);
    fhc[idx] = c; fhs[idx] = s; fhns[idx] = -s;
  }
  { // inverse row DFT: if*[hh][j] = e^{+i th}/256
    int hh = idx >> 5, j = idx & 31;
    int x = (j < 16) ? j : (224 + j);
    float th = STEP * (float)((x * hh) & 255);
    float c = cosf(th) * (1.f / 256.f), s = sinf(th) * (1.f / 256.f);
    ifc[idx] = c; ifs[idx] = s; ifns[idx] = -s;
  }
  { // inverse col (irfft hermitian fold): itab[2y+ri][ww]
    int n = idx >> 8, ww = idx & 255, y = n >> 1;
    float ay = ((y == 0) ? 1.f : 2.f) * (1.f / 256.f);
    float th = STEP * (float)((y * ww) & 255);
    itab[idx] = (n & 1) ? (-ay * sinf(th)) : (ay * cosf(th));
  }
}

// ---------------- E = (I + kw)^6 - I, per mode (32x32 complex) -------
__global__ void k_epow(const float* __restrict__ kw_r, const float* __restrict__ kw_i,
                       float* __restrict__ Er, float* __restrict__ Ei) {
  __shared__ float Ar[32][33], Ai[32][33], Br[32][33], Bi[32][33];
  int f = threadIdx.x, t = threadIdx.y, m = blockIdx.x;
  int x = m >> 4, y = m & 15;
  size_t kidx = ((size_t)(t * 32 + f) * 16 + x) * 16 + y;
  Ar[t][f] = kw_r[kidx] + ((t == f) ? 1.f : 0.f);      // A = I + M
  Ai[t][f] = kw_i[kidx];
  __syncthreads();
  float qr = 0.f, qi = 0.f;                            // Q = A*A
  for (int k = 0; k < 32; k++) {
    qr += Ar[t][k] * Ar[k][f] - Ai[t][k] * Ai[k][f];
    qi += Ar[t][k] * Ai[k][f] + Ai[t][k] * Ar[k][f];
  }
  __syncthreads();
  Ar[t][f] = qr; Ai[t][f] = qi;                        // A = Q
  __syncthreads();
  float rr = 0.f, ri = 0.f;                            // B = Q*Q
  for (int k = 0; k < 32; k++) {
    rr += Ar[t][k] * Ar[k][f] - Ai[t][k] * Ai[k][f];
    ri += Ar[t][k] * Ai[k][f] + Ai[t][k] * Ar[k][f];
  }
  Br[t][f] = rr; Bi[t][f] = ri;
  __syncthreads();
  float er = 0.f, ei = 0.f;                            // E = B*A = Q^3
  for (int k = 0; k < 32; k++) {
    er += Br[t][k] * Ar[k][f] - Bi[t][k] * Ai[k][f];
    ei += Br[t][k] * Ai[k][f] + Bi[t][k] * Ar[k][f];
  }
  Er[(size_t)m * 1024 + t * 32 + f] = er - ((t == f) ? 1.f : 0.f);
  Ei[(size_t)m * 1024 + t * 32 + f] = ei;
}

// ---------------- encoder + reconstruction branch --------------------
__global__ void k_encode(const float* __restrict__ x,
                         const float* __restrict__ enc_w, const float* __restrict__ enc_b,
                         const float* __restrict__ dec_w, const float* __restrict__ dec_b,
                         float* __restrict__ h0, float* __restrict__ rec) {
  __shared__ float sew[OPN * TN], seb[OPN], sdw[TN * OPN], sdb[TN];
  int tid = threadIdx.x;
  for (int i = tid; i < OPN * TN; i += blockDim.x) { sew[i] = enc_w[i]; sdw[i] = dec_w[i]; }
  if (tid < OPN) seb[tid] = enc_b[tid];
  if (tid < TN)  sdb[tid] = dec_b[tid];
  __syncthreads();
  int p = blockIdx.x * blockDim.x + tid;
  int off = p & (PSZ - 1);
  int bc = p >> 16, b = bc >> 1, c = bc & 1;
  float xi[TN];
#pragma unroll
  for (int t = 0; t < TN; t++)
    xi[t] = x[(((size_t)(b * TN + t) * CN + c) << 16) + off];
  float lat[OPN];
#pragma unroll
  for (int f = 0; f < OPN; f++) {
    float s = seb[f];
#pragma unroll
    for (int t = 0; t < TN; t++) s += xi[t] * sew[f * TN + t];
    lat[f] = tanhf(s);
    h0[((size_t)(bc * OPN + f) << 16) + off] = lat[f];
  }
#pragma unroll
  for (int t = 0; t < TN; t++) {
    float r = sdb[t];
#pragma unroll
    for (int f = 0; f < OPN; f++) r += sdw[t * OPN + f] * lat[f];
    rec[(((size_t)(b * TN + t) * CN + c) << 16) + off] = r;
  }
}

// ---------------- stage A: column DFT, G = h * Fw --------------------
// A-panel (16x256 fp32) is DMA'd into LDS by the Tensor Data Mover as two
// 16x128 half-tiles, double-buffered against WMMA via S_WAIT_TENSORCNT.
__global__ __launch_bounds__(32) void k_coldft(const float* __restrict__ h0,
                                               const float* __restrict__ fw,
                                               float* __restrict__ Gr, float* __restrict__ Gi) {
  __shared__ float sA[2 * 16 * 128];      // two half-tiles, 8 KB each
  int lane = threadIdx.x;
  int plane = blockIdx.x, m0 = blockIdx.y << 4;
  const float* A = h0 + ((size_t)plane << 16) + (size_t)m0 * 256;

  // LDS byte offset = low 32 bits of flat shared address (aperture mapping).
  // Escape the address through asm so the TDM writes stay visible to the
  // optimizer (prevents folding the ds reads of "never-stored" LDS).
  unsigned long long sAflat = (unsigned long long)(uintptr_t)&sA[0];
  asm volatile("" : "+v"(sAflat) :: "memory");
  unsigned lds0 = (unsigned)sAflat;

  tdm_load_2d(lds0,        A,       128, 16, 256);   // half 0: cols   0..127
  tdm_load_2d(lds0 + 8192, A + 128, 128, 16, 256);   // half 1: cols 128..255

  v8f a0 = {0.f,0.f,0.f,0.f,0.f,0.f,0.f,0.f};
  v8f a1 = {0.f,0.f,0.f,0.f,0.f,0.f,0.f,0.f};
  int arow = (lane & 15) * 128 + 2 * (lane >> 4);

  __builtin_amdgcn_s_wait_tensorcnt((unsigned short)1);  // half 0 landed (in-order)
  asm volatile("" ::: "memory");
  for (int k0 = 0; k0 < 128; k0 += 4) {
    v2f a; a.x = sA[arow + k0]; a.y = sA[arow + k0 + 1];
    v2f b0 = ldB(fw, 32, k0, 0, lane);
    v2f b1 = ldB(fw, 32, k0, 16, lane);
    a0 = wmma4(a, b0, a0);
    a1 = wmma4(a, b1, a1);
  }
  __builtin_amdgcn_s_wait_tensorcnt((unsigned short)0);  // half 1 landed
  asm volatile("" ::: "memory");
  for (int k0 = 128; k0 < 256; k0 += 4) {
    v2f a; a.x = sA[2048 + arow + k0 - 128]; a.y = sA[2048 + arow + k0 - 127];
    v2f b0 = ldB(fw, 32, k0, 0, lane);
    v2f b1 = ldB(fw, 32, k0, 16, lane);
    a0 = wmma4(a, b0, a0);
    a1 = wmma4(a, b1, a1);
  }

  int lrow = lane & 15, half = lane >> 4;
#pragma unroll
  for (int j = 0; j < 8; j++) {
    int row = m0 + j + 8 * half;
    { int n = lrow; float* dst = (n & 1) ? Gi : Gr;
      dst[(size_t)plane * 4096 + row * 16 + (n >> 1)] = a0[j]; }
    { int n = 16 + lrow; float* dst = (n & 1) ? Gi : Gr;
      dst[(size_t)plane * 4096 + row * 16 + (n >> 1)] = a1[j]; }
  }
}

// ---------------- stage B: row DFT to 32 kept modes ------------------
__global__ __launch_bounds__(32) void k_rowdft(const float* __restrict__ Gr,
                                               const float* __restrict__ Gi,
                                               const float* __restrict__ fhc,
                                               const float* __restrict__ fhs,
                                               const float* __restrict__ fhns,
                                               float* __restrict__ Tr, float* __restrict__ Ti) {
  int lane = threadIdx.x;
  int plane = blockIdx.x;
  const float* gr = Gr + (size_t)plane * 4096;
  const float* gi = Gi + (size_t)plane * 4096;
  v8f tr0 = {0.f,0.f,0.f,0.f,0.f,0.f,0.f,0.f}, tr1 = tr0, ti0 = tr0, ti1 = tr0;
  for (int k0 = 0; k0 < 256; k0 += 4) {
    __builtin_prefetch(gr + (k0 + 16) * 16, 0, 1);
    v2f br = ldB(gr, 16, k0, 0, lane);
    v2f bi = ldB(gi, 16, k0, 0, lane);
    v2f c0 = ldA(fhc, 256, 0, k0, lane),  c1 = ldA(fhc, 256, 16, k0, lane);
    v2f s0 = ldA(fhs, 256, 0, k0, lane),  s1 = ldA(fhs, 256, 16, k0, lane);
    v2f n0 = ldA(fhns, 256, 0, k0, lane), n1 = ldA(fhns, 256, 16, k0, lane);
    // T = e^{-i th} * G : Tr = cos*Gr + sin*Gi ; Ti = cos*Gi - sin*Gr
    tr0 = wmma4(c0, br, tr0); tr0 = wmma4(s0, bi, tr0);
    ti0 = wmma4(c0, bi, ti0); ti0 = wmma4(n0, br, ti0);
    tr1 = wmma4(c1, br, tr1); tr1 = wmma4(s1, bi, tr1);
    ti1 = wmma4(c1, bi, ti1); ti1 = wmma4(n1, br, ti1);
  }
  int lrow = lane & 15, half = lane >> 4;
#pragma unroll
  for (int j = 0; j < 8; j++) {
    int r0 = j + 8 * half, r1 = 16 + r0;
    Tr[(size_t)plane * 512 + r0 * 16 + lrow] = tr0[j];
    Ti[(size_t)plane * 512 + r0 * 16 + lrow] = ti0[j];
    Tr[(size_t)plane * 512 + r1 * 16 + lrow] = tr1[j];
    Ti[(size_t)plane * 512 + r1 * 16 + lrow] = ti1[j];
  }
}

// ---------------- stage C: per-mode 32x32 complex mix with E ---------
__global__ __launch_bounds__(32) void k_modemix(const float* __restrict__ Tr,
                                                const float* __restrict__ Ti,
                                                const float* __restrict__ Er,
                                                const float* __restrict__ Ei,
                                                float* __restrict__ Ur, float* __restrict__ Ui) {
  __shared__ float str[32], sti[32];
  int id = blockIdx.x;
  int bc = id >> 9, jm = id & 511, j = jm >> 4, y = jm & 15;
  int f = threadIdx.x;
  str[f] = Tr[((size_t)(bc * 32 + f) * 32 + j) * 16 + y];
  sti[f] = Ti[((size_t)(bc * 32 + f) * 32 + j) * 16 + y];
  __syncthreads();
  const float* er = Er + (size_t)((j & 15) * 16 + y) * 1024 + f;  // same E for top/bottom rows
  const float* ei = Ei + (size_t)((j & 15) * 16 + y) * 1024 + f;
  float ur = 0.f, ui = 0.f;
  for (int k = 0; k < 32; k++) {
    float a = str[k], b = sti[k], c = er[k * 32], d = ei[k * 32];
    ur += a * c - b * d;
    ui += a * d + b * c;
  }
  Ur[((size_t)(bc * 32 + f) * 32 + j) * 16 + y] = ur;
  Ui[((size_t)(bc * 32 + f) * 32 + j) * 16 + y] = ui;
}

// ---------------- stage D: inverse row DFT ---------------------------
__global__ __launch_bounds__(32) void k_irow(const float* __restrict__ Ur,
                                             const float* __restrict__ Ui,
                                             const float* __restrict__ ifc,
                                             const float* __restrict__ ifs,
                                             const float* __restrict__ ifns,
                                             float* __restrict__ Z) {
  int lane = threadIdx.x;
  int plane = blockIdx.x, m0 = blockIdx.y << 4;
  const float* ur = Ur + (size_t)plane * 512;
  const float* ui = Ui + (size_t)plane * 512;
  v8f zr = {0.f,0.f,0.f,0.f,0.f,0.f,0.f,0.f}, zi = zr;
  for (int k0 = 0; k0 < 32; k0 += 4) {
    v2f br = ldB(ur, 16, k0, 0, lane);
    v2f bi = ldB(ui, 16, k0, 0, lane);
    v2f c  = ldA(ifc, 32, m0, k0, lane);
    v2f s  = ldA(ifs, 32, m0, k0, lane);
    v2f ns = ldA(ifns, 32, m0, k0, lane);
    // Z = e^{+i th}/256 * U : Zr = c*Ur - s*Ui ; Zi = c*Ui + s*Ur
    zr = wmma4(c, br, zr); zr = wmma4(ns, bi, zr);
    zi = wmma4(c, bi, zi); zi = wmma4(s, br, zi);
  }
  int lrow = lane & 15, half = lane >> 4;
#pragma unroll
  for (int j = 0; j < 8; j++) {
    int row = m0 + j + 8 * half;
    Z[(size_t)plane * 8192 + row * 32 + 2 * lrow]     = zr[j];
    Z[(size_t)plane * 8192 + row * 32 + 2 * lrow + 1] = zi[j];
  }
}

// ---------------- stage E: inverse col DFT + residual add ------------
__global__ __launch_bounds__(32) void k_icol(const float* __restrict__ Z,
                                             const float* __restrict__ itab,
                                             const float* __restrict__ h0,
                                             float* __restrict__ h6) {
  int lane = threadIdx.x;
  int plane = blockIdx.x, m0 = blockIdx.y << 4, n0 = blockIdx.z << 5;
  const float* A = Z + (size_t)plane * 8192;
  v8f a0 = {0.f,0.f,0.f,0.f,0.f,0.f,0.f,0.f}, a1 = a0;
  for (int k0 = 0; k0 < 32; k0 += 4) {
    v2f a  = ldA(A, 32, m0, k0, lane);
    v2f b0 = ldB(itab, 256, k0, n0, lane);
    v2f b1 = ldB(itab, 256, k0, n0 + 16, lane);
    a0 = wmma4(a, b0, a0);
    a1 = wmma4(a, b1, a1);
  }
  int lrow = lane & 15, half = lane >> 4;
#pragma unroll
  for (int j = 0; j < 8; j++) {
    int row = m0 + j + 8 * half;
    const float* hp = h0 + ((size_t)plane << 16) + (size_t)row * 256;
    float* dp = h6 + ((size_t)plane << 16) + (size_t)row * 256;
    dp[n0 + lrow]      = a0[j] + hp[n0 + lrow];       // h6 = (I+K)^6 h0
    dp[n0 + 16 + lrow] = a1[j] + hp[n0 + 16 + lrow];
  }
}

// ---------------- epilogue: w0 skip + tanh + 0.1 + decoder -----------
__global__ void k_epilogue(const float* __restrict__ h0, const float* __restrict__ h6,
                           const float* __restrict__ w0_w, const float* __restrict__ w0_b,
                           const float* __restrict__ dec_w, const float* __restrict__ dec_b,
                           float* __restrict__ out) {
  __shared__ float sw0[1024], sb0[32], sdw[320], sdb[10];
  int tid = threadIdx.x;
  for (int i = tid; i < 1024; i += blockDim.x) sw0[i] = w0_w[i];
  for (int i = tid; i < 320; i += blockDim.x) sdw[i] = dec_w[i];
  if (tid < 32) sb0[tid] = w0_b[tid];
  if (tid < 10) sdb[tid] = dec_b[tid];
  __syncthreads();
  int p = blockIdx.x * blockDim.x + tid;
  int off = p & (PSZ - 1);
  int bc = p >> 16, b = bc >> 1, c = bc & 1;
  float a[32], v[32];
#pragma unroll
  for (int f = 0; f < 32; f++) {
    a[f] = h0[((size_t)(bc * 32 + f) << 16) + off];
    v[f] = h6[((size_t)(bc * 32 + f) << 16) + off];
  }
#pragma unroll
  for (int f = 0; f < 32; f++) {
    float s = sb0[f];
#pragma unroll
    for (int k = 0; k < 32; k++) s += sw0[f * 32 + k] * a[k];
    v[f] = tanhf(s + v[f]) * 0.1f;
  }
#pragma unroll
  for (int t = 0; t < 10; t++) {
    float o = sdb[t];
#pragma unroll
    for (int f = 0; f < 32; f++) o += sdw[t * 32 + f] * v[f];
    out[(((size_t)(b * 10 + t) * 2 + c) << 16) + off] = o;
  }
}

// ---------------- host ----------------
extern "C" void kernel_launch(void* const* d_in, const int* in_sizes, int n_in,
                              void* d_out, int out_size, void* d_ws, size_t ws_size,
                              hipStream_t stream) {
  const float* x     = (const float*)d_in[0];
  const float* enc_w = (const float*)d_in[1];
  const float* enc_b = (const float*)d_in[2];
  const float* dec_w = (const float*)d_in[3];
  const float* dec_b = (const float*)d_in[4];
  const float* kw_r  = (const float*)d_in[5];
  const float* kw_i  = (const float*)d_in[6];
  const float* w0_w  = (const float*)d_in[7];
  const float* w0_b  = (const float*)d_in[8];
  float* out = (float*)d_out;
  float* rec = out + OUTSZ;

  float* ws   = (float*)d_ws;          // needs ~309 MB
  float* h0   = ws + OFF_H0;
  float* h6   = ws + OFF_H6;
  float* Gr   = ws + OFF_GR;
  float* Gi   = ws + OFF_GI;
  float* Trb  = ws + OFF_TR;
  float* Tib  = ws + OFF_TI;
  float* Urb  = ws + OFF_UR;
  float* Uib  = ws + OFF_UI;
  float* Zb   = ws + OFF_Z;
  float* Er   = ws + OFF_ER;
  float* Ei   = ws + OFF_EI;
  float* fw   = ws + OFF_FW;
  float* fhc  = ws + OFF_FHC;
  float* fhs  = ws + OFF_FHS;
  float* fhns = ws + OFF_FHNS;
  float* ifc  = ws + OFF_IFC;
  float* ifs  = ws + OFF_IFS;
  float* ifns = ws + OFF_IFNS;
  float* itab = ws + OFF_ITAB;

  k_tables<<<32, 256, 0, stream>>>(fw, fhc, fhs, fhns, ifc, ifs, ifns, itab);
  k_epow<<<256, dim3(32, 32), 0, stream>>>(kw_r, kw_i, Er, Ei);
  k_encode<<<NPIX / 256, 256, 0, stream>>>(x, enc_w, enc_b, dec_w, dec_b, h0, rec);
  k_coldft<<<dim3(NPLANE, 16), 32, 0, stream>>>(h0, fw, Gr, Gi);
  k_rowdft<<<NPLANE, 32, 0, stream>>>(Gr, Gi, fhc, fhs, fhns, Trb, Tib);
  k_modemix<<<BCN * 512, 32, 0, stream>>>(Trb, Tib, Er, Ei, Urb, Uib);
  k_irow<<<dim3(NPLANE, 16), 32, 0, stream>>>(Urb, Uib, ifc, ifs, ifns, Zb);
  k_icol<<<dim3(NPLANE, 16, 8), 32, 0, stream>>>(Zb, itab, h0, h6);
  k_epilogue<<<NPIX / 256, 256, 0, stream>>>(h0, h6, w0_w, w0_b, dec_w, dec_b, out);
}